// SampleAndGroup_61486751810144
// MI455X (gfx1250) — compile-verified
//
#include <hip/hip_runtime.h>

#define BATCH   8
#define NPTS    8192
#define NPOINT  1024
#define NSAMPLE 32
#define CIN     32
#define CH      35            // 3 (grouped_xyz) + 32 (grouped_points)
#define RADIUS2 0.0625f       // 0.25^2
#define BLK2    512           // 16 waves of 32

typedef __attribute__((ext_vector_type(2))) float v2f;
typedef __attribute__((ext_vector_type(8))) float v8f;
typedef __attribute__((ext_vector_type(4))) unsigned int u32x4;
typedef __attribute__((ext_vector_type(8))) int          i32x8;
typedef __attribute__((ext_vector_type(4))) int          i32x4;

// LDS byte offset of a generic pointer into shared memory: the flat-LDS
// aperture keeps the workgroup-relative LDS offset in addr[31:0] (ISA 10.2).
__device__ __forceinline__ unsigned lds_off(const void* p) {
  return (unsigned)(uintptr_t)p;
}

// Tensor Data Mover: DMA one contiguous run of `n4B` DWORDs from global
// memory into LDS (1-row 2D tile, D# per ISA 8.3/8.4). Issue from ONE wave
// only (TDM ignores EXEC), then s_wait_tensorcnt 0 before consuming.
__device__ __forceinline__ void tdm_load_1d(unsigned dst_lds_byte,
                                            const void* src_global,
                                            unsigned n4B) {
  unsigned long long ga = (unsigned long long)(uintptr_t)src_global;
  u32x4 g0;
  g0[0] = 1u;                                    // count=1, user descriptor
  g0[1] = dst_lds_byte;                          // lds_addr  [63:32]
  g0[2] = (unsigned)ga;                          // global_addr[31:0]
  g0[3] = (unsigned)((ga >> 32) & 0x01FFFFFFu)   // global_addr[56:32]
        | (2u << 30);                            // type=2 ("image")
  i32x8 g1;
  g1[0] = (int)(2u << 16);                       // data_size=4B; no multicast
  g1[1] = (int)((n4B & 0xffffu) << 16);          // tensor_dim0 [79:48] lo
  g1[2] = (int)((n4B >> 16) & 0xffffu)           // tensor_dim0 hi
        | (1 << 16);                             // tensor_dim1 = 1
  g1[3] = (int)((n4B & 0xffffu) << 16);          // tile_dim0 [127:112]
  g1[4] = 1;                                     // tile_dim1 = 1
  g1[5] = (int)n4B;                              // tensor_dim0_stride lo
  g1[6] = 0;
  g1[7] = 0;
  i32x4 z4 = {0, 0, 0, 0};
#if defined(__clang_major__) && __clang_major__ >= 23
  i32x8 z8 = {0, 0, 0, 0, 0, 0, 0, 0};
  __builtin_amdgcn_tensor_load_to_lds(g0, g1, z4, z4, z8, 0);
#else
  __builtin_amdgcn_tensor_load_to_lds(g0, g1, z4, z4, 0);
#endif
}

// --------------------------------------------------------------------------
// Kernel 1: furthest point sampling. One block per batch. Batch coords are
// DMA'd into LDS by the TDM, then repacked as [x,y,z,dist] float4 so the
// 1023 serially dependent scan iterations cost one ds_load_b128 + one
// ds_store_b32 per point.
// --------------------------------------------------------------------------
__global__ __launch_bounds__(1024, 1)
void fps_kernel(const float* __restrict__ xyz, float* __restrict__ new_xyz) {
  extern __shared__ float smem[];
  float* buf = smem;                 // NPTS*4 : x,y,z,dist interleaved
  float* aos = buf + NPTS * 4;       // NPTS*3 : TDM staging (row-major xyz)
  __shared__ float redV[32];
  __shared__ int   redI[32];
  __shared__ float curP[3];

  const int b    = blockIdx.x;
  const int tid  = threadIdx.x;
  const int lane = tid & 31;
  const int wid  = tid >> 5;
  const float* base = xyz + (size_t)b * NPTS * 3;

  if (wid == 0) {                    // single wave issues the DMA
    tdm_load_1d(lds_off(aos), base, NPTS * 3);
    __builtin_amdgcn_s_wait_tensorcnt(0);
  }
  __syncthreads();

  for (int j = tid; j < NPTS; j += 1024) {
    float4 v;
    v.x = aos[3 * j + 0];
    v.y = aos[3 * j + 1];
    v.z = aos[3 * j + 2];
    v.w = 3.0e38f;                   // running min distance
    *(float4*)(buf + 4 * j) = v;
  }
  if (tid == 0) { curP[0] = aos[0]; curP[1] = aos[1]; curP[2] = aos[2]; }
  __syncthreads();

  float cx = curP[0], cy = curP[1], cz = curP[2];

  for (int i = 0; i < NPOINT; ++i) {
    if (tid == 0) {
      float* o = new_xyz + ((size_t)b * NPOINT + i) * 3;
      o[0] = cx; o[1] = cy; o[2] = cz;
    }
    if (i == NPOINT - 1) break;

    float bestV = -1.0f; int bestI = NPTS;
    for (int j = tid; j < NPTS; j += 1024) {
      float4 v = *(const float4*)(buf + 4 * j);
      float dx = v.x - cx, dy = v.y - cy, dz = v.z - cz;
      float d2 = dx * dx + dy * dy + dz * dz;
      float d  = fminf(v.w, d2);
      buf[4 * j + 3] = d;
      if (d > bestV) { bestV = d; bestI = j; }
    }
    #pragma unroll
    for (int off = 16; off > 0; off >>= 1) {
      float ov = __shfl_down(bestV, off, 32);
      int   oi = __shfl_down(bestI, off, 32);
      if (ov > bestV || (ov == bestV && oi < bestI)) { bestV = ov; bestI = oi; }
    }
    if (lane == 0) { redV[wid] = bestV; redI[wid] = bestI; }
    __syncthreads();
    if (wid == 0) {
      bestV = redV[lane]; bestI = redI[lane];
      #pragma unroll
      for (int off = 16; off > 0; off >>= 1) {
        float ov = __shfl_down(bestV, off, 32);
        int   oi = __shfl_down(bestI, off, 32);
        if (ov > bestV || (ov == bestV && oi < bestI)) { bestV = ov; bestI = oi; }
      }
      if (lane == 0) {
        curP[0] = buf[4 * bestI + 0];
        curP[1] = buf[4 * bestI + 1];
        curP[2] = buf[4 * bestI + 2];
      }
    }
    __syncthreads();
    cx = curP[0]; cy = curP[1]; cz = curP[2];
  }
}

// --------------------------------------------------------------------------
// Kernel 2: ball query (WMMA pairwise distances -> in-radius bitmask ->
// rank-scan selection) fused with grouping. One block = 16 centers of one
// batch; 16 waves each cover a 512-point slice with V_WMMA_F32_16X16X4_F32.
//   d2 = |c|^2 + |p|^2 - 2 c.p :  A = (-2cx,-2cy,-2cz,0), B = (px,py,pz,0),
//   C preloaded with |c|^2 + |p|^2; one WMMA per 16x16 tile.
// --------------------------------------------------------------------------
__global__ __launch_bounds__(BLK2, 1)
void group_kernel(const float* __restrict__ xyz,
                  const float* __restrict__ points,
                  const float* __restrict__ new_xyz,
                  float* __restrict__ new_points) {
  extern __shared__ float smem[];
  float*    xs   = smem;                         // NPTS
  float*    ys   = xs + NPTS;                    // NPTS
  float*    zs   = ys + NPTS;                    // NPTS
  float*    pn   = zs + NPTS;                    // NPTS  |p|^2
  float*    aos  = pn + NPTS;                    // NPTS*3 TDM staging
  unsigned* mask = (unsigned*)(aos + NPTS * 3);  // 16 rows x 256 words
  float*    cxs  = (float*)(mask + 16 * 256);    // 16
  float*    cys  = cxs + 16;
  float*    czs  = cys + 16;
  float*    cns  = czs + 16;                     // |c|^2
  int*      idxs = (int*)(cns + 16);             // 16 x NSAMPLE
  int*      cnts = idxs + 16 * NSAMPLE;          // 16

  const int blk  = blockIdx.x;
  const int b    = blk >> 6;                     // NPOINT/16 == 64 tiles/batch
  const int m0   = (blk & 63) * 16;
  const int tid  = threadIdx.x;
  const int lane = tid & 31;
  const int wid  = tid >> 5;

  const float* xbase = xyz + (size_t)b * NPTS * 3;

  // ---- Phase 0: TDM-stage batch coords; centers + L2 warm in parallel ----
  if (wid == 0) {                    // single wave issues the DMA
    tdm_load_1d(lds_off(aos), xbase, NPTS * 3);
    __builtin_amdgcn_s_wait_tensorcnt(0);
  }
  if (tid >= 32 && tid < 48) {       // wave 1: the 16 centers
    const int m = tid - 32;
    const float* c = new_xyz + ((size_t)b * NPOINT + m0 + m) * 3;
    float x = c[0], y = c[1], z = c[2];
    cxs[m] = x; cys[m] = y; czs[m] = z; cns[m] = x * x + y * y + z * z;
  }
  if (wid >= 2) {                    // waves 2..15: warm L2 for the gather
    const char* pbase = (const char*)(points + (size_t)b * NPTS * CIN);
    for (int j = tid - 64; j < NPTS / 2; j += BLK2 - 64)
      __builtin_prefetch(pbase + (size_t)j * 256, 0, 1);
  }
  __syncthreads();

  // AoS -> SoA + norms (LDS to LDS)
  for (int j = tid; j < NPTS; j += BLK2) {
    float x = aos[3 * j + 0], y = aos[3 * j + 1], z = aos[3 * j + 2];
    xs[j] = x; ys[j] = y; zs[j] = z; pn[j] = x * x + y * y + z * z;
  }
  __syncthreads();

  // ---- Phase 1: WMMA distance tiles -> radius bitmask in LDS -------------
  {
    const int half = lane >> 4;       // lane half selects K pair (0/1 vs 2/3)
    const int l16  = lane & 15;
    v2f A;                            // A 16x4: V0 = K0|K2, V1 = K1|K3
    A.x = half ? (-2.0f * czs[l16]) : (-2.0f * cxs[l16]);
    A.y = half ? 0.0f               : (-2.0f * cys[l16]);
    float cn[8];                      // |c|^2 for this lane-half's 8 rows
    #pragma unroll
    for (int r = 0; r < 8; ++r) cn[r] = cns[(half ? 8 : 0) + r];

    const float* bxs = half ? zs : xs;  // branch-free: per-lane base pointer
    const float yscl = half ? 0.0f : 1.0f;

    const int nbeg = wid * (NPTS / 16);           // 512 points per wave
    for (int n0 = nbeg; n0 < nbeg + NPTS / 16; n0 += 32) {
      const int na = n0 + l16, nb = n0 + 16 + l16;
      v2f Ba, Bb;                     // B 4x16: V0 = K0|K2, V1 = K1|K3
      Ba.x = bxs[na];  Ba.y = ys[na] * yscl;
      Bb.x = bxs[nb];  Bb.y = ys[nb] * yscl;
      const float pa = pn[na], pb = pn[nb];
      v8f Ca, Cb;
      #pragma unroll
      for (int r = 0; r < 8; ++r) { Ca[r] = cn[r] + pa; Cb[r] = cn[r] + pb; }
      Ca = __builtin_amdgcn_wmma_f32_16x16x4_f32(false, A, false, Ba,
                                                 (short)0, Ca, false, false);
      Cb = __builtin_amdgcn_wmma_f32_16x16x4_f32(false, A, false, Bb,
                                                 (short)0, Cb, false, false);
      // D layout: lanes 0-15 = row r, lanes 16-31 = row r+8, col = lane&15.
      // Funnel each row's 32-bit mask word into the lane that stores it
      // (lane r -> row r, lane r+8 -> row r+8): v_cndmask, no exec churn.
      unsigned myw = 0;
      #pragma unroll
      for (int r = 0; r < 8; ++r) {
        unsigned ba = __builtin_amdgcn_ballot_w32(Ca[r] < RADIUS2);
        unsigned bb = __builtin_amdgcn_ballot_w32(Cb[r] < RADIUS2);
        unsigned wlo = (ba & 0xffffu) | (bb << 16);         // row r
        unsigned whi = (ba >> 16)     | (bb & 0xffff0000u); // row r+8
        myw = (lane == r)     ? wlo : myw;
        myw = (lane == r + 8) ? whi : myw;
      }
      const int wix = n0 >> 5;
      if (lane < 16) mask[lane * 256 + wix] = myw;
    }
  }
  __syncthreads();

  // ---- Phase 2: per-center selection of first NSAMPLE in-radius indices --
  {
    const unsigned* mrow = mask + wid * 256;      // wave w handles center w
    unsigned mw[8]; int pc = 0;
    #pragma unroll
    for (int k = 0; k < 8; ++k) { mw[k] = mrow[lane * 8 + k]; pc += __popc(mw[k]); }
    int incl = pc;
    #pragma unroll
    for (int d = 1; d < 32; d <<= 1) {
      int o = __shfl_up(incl, d, 32);
      if (lane >= d) incl += o;
    }
    int rank  = incl - pc;                        // exclusive prefix = bit rank
    int total = __shfl(incl, 31, 32);
    int* row = idxs + wid * NSAMPLE;
    #pragma unroll
    for (int k = 0; k < 8; ++k) {
      unsigned wv = mw[k];
      while (wv) {
        int bit = __ffs(wv) - 1;
        wv &= wv - 1;
        if (rank < NSAMPLE) row[rank] = ((lane * 8 + k) << 5) + bit;
        ++rank;
      }
    }
    if (lane == 0) cnts[wid] = total < NSAMPLE ? total : NSAMPLE;
  }
  __syncthreads();

  // pad unfilled slots with the first found index (slot 0 always exists:
  // each center is itself a sampled point, so d2 == 0 < r^2)
  {
    const int m = tid >> 5, s = tid & 31;
    const int c = cnts[m];
    const int fill = idxs[m * NSAMPLE];
    if (s >= c) idxs[m * NSAMPLE + s] = fill;
  }
  __syncthreads();

  // ---- Phase 3: gather + center-subtract + concat writeout ---------------
  {
    const int m = tid >> 5, s = tid & 31;
    const int gi = idxs[m * NSAMPLE + s];
    float* o = new_points + (((size_t)b * NPOINT + m0 + m) * NSAMPLE + s) * CH;
    o[0] = xs[gi] - cxs[m];
    o[1] = ys[gi] - cys[m];
    o[2] = zs[gi] - czs[m];
    const float* prow = points + ((size_t)b * NPTS + gi) * CIN;  // 128B aligned
    #pragma unroll
    for (int c4 = 0; c4 < CIN; c4 += 4) {
      float4 v = *(const float4*)(prow + c4);
      o[3 + c4 + 0] = v.x; o[3 + c4 + 1] = v.y;
      o[3 + c4 + 2] = v.z; o[3 + c4 + 3] = v.w;
    }
  }
}

// --------------------------------------------------------------------------
extern "C" void kernel_launch(void* const* d_in, const int* in_sizes, int n_in,
                              void* d_out, int out_size, void* d_ws, size_t ws_size,
                              hipStream_t stream) {
  (void)in_sizes; (void)n_in; (void)out_size; (void)d_ws; (void)ws_size;
  const float* xyz    = (const float*)d_in[0];   // (8, 8192, 3)
  const float* points = (const float*)d_in[1];   // (8, 8192, 32)
  float* out        = (float*)d_out;
  float* new_xyz    = out;                                   // (8,1024,3)
  float* new_points = out + (size_t)BATCH * NPOINT * 3;      // (8,1024,32,35)

  const size_t sh1 = (size_t)NPTS * 4 * sizeof(float)        // [x,y,z,dist]
                   + (size_t)NPTS * 3 * sizeof(float);       // TDM staging
  fps_kernel<<<dim3(BATCH), dim3(1024), sh1, stream>>>(xyz, new_xyz);

  const size_t sh2 = (size_t)4 * NPTS * sizeof(float)        // SoA + norms
                   + (size_t)NPTS * 3 * sizeof(float)        // TDM staging
                   + (size_t)16 * 256 * sizeof(unsigned)     // radius bitmask
                   + (size_t)4 * 16 * sizeof(float)          // centers + |c|^2
                   + (size_t)16 * NSAMPLE * sizeof(int)      // selected idx
                   + (size_t)16 * sizeof(int);               // counts
  group_kernel<<<dim3(BATCH * (NPOINT / 16)), dim3(BLK2), sh2, stream>>>(
      xyz, points, new_xyz, new_points);
}